// Sparse_Conv_84731114815960
// MI455X (gfx1250) — compile-verified
//
#include <hip/hip_runtime.h>

typedef float v2f __attribute__((ext_vector_type(2)));
typedef float v8f __attribute__((ext_vector_type(8)));

#define B_    8
#define C_    64
#define H_    96
#define W_    96
#define OC_   64
#define SUB_  16
#define OH_   94
#define OW_   94
#define XPAD  34                         // padded channel dim (32 used) -> even, bank-friendly
#define XROWS 6                          // 4 output rows need 6 input rows
#define XBUF_FLOATS (XROWS * W_ * XPAD)  // 19584 floats
#define NSTEP 72                         // 2 ch-tiles * 9 taps * 4 K-chunks
#define APACK_FLOATS (NSTEP * 64)        // 4608 floats
#define SMEM_BYTES ((XBUF_FLOATS + APACK_FLOATS) * 4)

// gfx1250 async DMA global->LDS (ASYNCcnt) if this toolchain exposes it.
#if defined(__has_builtin)
#if __has_builtin(__builtin_amdgcn_global_load_async_to_lds_b32) && \
    __has_builtin(__builtin_amdgcn_s_wait_asynccnt)
#define USE_ASYNC_LDS 1
#endif
#endif

typedef __attribute__((address_space(1))) int gint;   // global (prints as __device__ int*)
typedef __attribute__((address_space(3))) int lint;   // LDS    (prints as __shared__ int*)

// out[b,o,y,x] = bias[o] + sum_{s<16,ky,kx} X[b,(o+s)&63,y+ky,x+kx] * W[o,s,ky,kx]
// Banded implicit GEMM on v_wmma_f32_16x16x4_f32.
__global__ __launch_bounds__(256) void sparse_conv_wmma(
    const float* __restrict__ X, const float* __restrict__ Wt,
    const float* __restrict__ bias, float* __restrict__ out)
{
  extern __shared__ float smem[];
  float* xbuf  = smem;                 // [row][x][u<34] : X tile, channel-interleaved
  float* apack = smem + XBUF_FLOATS;   // [step][lane][2] : A operands in WMMA lane layout

  const int tid  = threadIdx.x;
  const int lane = tid & 31;           // wave32
  const int wave = tid >> 5;
  const int j    = lane & 15;          // N (x within tile)
  const int lhi  = lane >> 4;          // K-pair / M-high selector

  const int blk = blockIdx.x;
  const int m0  = (blk & 3) * 16;      // output-channel tile
  const int yb  = (blk >> 2) % 24;     // 4-row block of output rows
  const int b   = blk / 96;            // batch
  const int y0  = yb * 4;

  // ---- Stage X tile: 32 logical channels (m0..m0+31 mod 64) x 6 rows x 96 cols ----
  // Thread tid owns channel u = tid>>3, x-phase sub = tid&7 (shift/add indexing only).
  {
    const int u   = tid >> 3;                 // 0..31 logical channel offset
    const int sub = tid & 7;
    const int c   = (m0 + u) & 63;
    const float* gch = X + ((size_t)(b * C_ + c)) * (H_ * W_);
#ifdef USE_ASYNC_LDS
    #pragma unroll
    for (int row = 0; row < XROWS; ++row) {
      int gy = y0 + row; if (gy > H_ - 1) gy = H_ - 1;   // clamp, values unused if OOB
      const float* grow_ = gch + gy * W_;
      #pragma unroll
      for (int m = 0; m < 12; ++m) {
        const int x = m * 8 + sub;                       // lanes -> contiguous 32B runs
        __builtin_amdgcn_global_load_async_to_lds_b32(
            (gint*)(const void*)(grow_ + x),
            (lint*)(void*)(xbuf + (row * W_ + x) * XPAD + u),
            0, 0);
      }
    }
#else
    #pragma unroll
    for (int row = 0; row < XROWS; ++row) {
      int gy = y0 + row; if (gy > H_ - 1) gy = H_ - 1;
      const float* grow_ = gch + gy * W_;
      #pragma unroll
      for (int m = 0; m < 3; ++m) {
        const int ch4 = m * 8 + sub;                     // float4 chunk along x
        const float4 v = *(const float4*)(grow_ + ch4 * 4);
        int base = (row * W_ + ch4 * 4) * XPAD + u;
        xbuf[base]            = v.x;
        xbuf[base +     XPAD] = v.y;
        xbuf[base + 2 * XPAD] = v.z;
        xbuf[base + 3 * XPAD] = v.w;
      }
    }
#endif
  }

  // ---- Build packed A (band weights) in WMMA A layout: lane = M + 16*(K/2), vgpr = K&1 ----
  // Overlaps with the in-flight async DMA above.
  {
    const int i   = lane & 15;                // M
    const int kk0 = (lane >> 4) << 1;         // K base for this lane
    for (int k = 0; k < 9; ++k) {
      const int step  = wave + 8 * k;         // 0..71, each (wave,k) unique
      const int chunk = step & 3;
      const int q     = step >> 2;            // (t*3+ky)*3+kx
      const int kx    = q % 3;
      const int ky    = (q / 3) % 3;
      const int t     = q / 9;
      float v0, v1;
      {
        int sidx = (16 * t + 4 * chunk + kk0 - i) & 63;
        v0 = (sidx < SUB_) ? Wt[(((m0 + i) * SUB_ + sidx) * 3 + ky) * 3 + kx] : 0.0f;
      }
      {
        int sidx = (16 * t + 4 * chunk + kk0 + 1 - i) & 63;
        v1 = (sidx < SUB_) ? Wt[(((m0 + i) * SUB_ + sidx) * 3 + ky) * 3 + kx] : 0.0f;
      }
      apack[step * 64 + lane * 2]     = v0;
      apack[step * 64 + lane * 2 + 1] = v1;
    }
  }

#ifdef USE_ASYNC_LDS
  __builtin_amdgcn_s_wait_asynccnt(0);        // my DMA landed in LDS
#endif
  __syncthreads();                            // publish to all waves

  // ---- 24 (row r, x-tile) 16x16 output tiles; each wave takes 3 ----
  for (int tile = wave; tile < 24; tile += 8) {
    const int r  = tile / 6;
    const int xt = tile % 6;
    const int x0 = xt * 16;

    v8f acc;
    #pragma unroll
    for (int vg = 0; vg < 8; ++vg)
      acc[vg] = bias[m0 + vg + 8 * lhi];      // C/D layout: M = vg + 8*lhi, N = j

    for (int t = 0; t < 2; ++t) {
      #pragma unroll
      for (int ky = 0; ky < 3; ++ky) {
        #pragma unroll
        for (int kx = 0; kx < 3; ++kx) {
          const int row = r + ky;
          const int xx  = x0 + kx + j;
          // B layout: lane holds K = 2*lhi + v at column N = j -> adjacent channels u, u+1
          const float* bptr = &xbuf[(row * W_ + xx) * XPAD + 16 * t + 2 * lhi];
          const int stepBase = ((t * 3 + ky) * 3 + kx) * 4;
          #pragma unroll
          for (int chunk = 0; chunk < 4; ++chunk) {
            v2f a  = *(const v2f*)&apack[(stepBase + chunk) * 64 + lane * 2]; // ds_load_b64
            v2f bv = *(const v2f*)(bptr + 4 * chunk);                        // ds_load_b64
            acc = __builtin_amdgcn_wmma_f32_16x16x4_f32(false, a, false, bv,
                                                        (short)0, acc, false, false);
          }
        }
      }
    }

    const int y = y0 + r;
    const int x = x0 + j;
    #pragma unroll
    for (int vg = 0; vg < 8; ++vg) {
      int M = vg + 8 * lhi;
      if (y < OH_ && x < OW_)
        out[(((size_t)b * OC_ + m0 + M) * OH_ + y) * OW_ + x] = acc[vg];
    }
  }
}

extern "C" void kernel_launch(void* const* d_in, const int* in_sizes, int n_in,
                              void* d_out, int out_size, void* d_ws, size_t ws_size,
                              hipStream_t stream) {
  const float* X  = (const float*)d_in[0];
  const float* Wt = (const float*)d_in[1];
  const float* bs = (const float*)d_in[2];
  float* out = (float*)d_out;
  dim3 grid(B_ * 24 * 4);   // batch * row-blocks * channel-tiles = 768 workgroups
  dim3 block(256);          // 8 wave32
  hipLaunchKernelGGL(sparse_conv_wmma, grid, block, SMEM_BYTES, stream,
                     X, Wt, bs, out);
}